// global_rel_trafo_25305947308336
// MI455X (gfx1250) — compile-verified
//
#include <hip/hip_runtime.h>

// ---------------------------------------------------------------------------
// Compile-time PGA G(3,0,1) Cayley tables (blade order matches gatr/gafl).
// Blades as bitmasks (bit g = generator e_g); e0^2 = 0, e1..e3 square to +1.
// ---------------------------------------------------------------------------
namespace pga {

constexpr int MASK[16] = {
    0b0000,                                  // scalar
    0b0001, 0b0010, 0b0100, 0b1000,          // e0 e1 e2 e3
    0b0011, 0b0101, 0b1001, 0b0110, 0b1010, 0b1100,  // e01 e02 e03 e12 e13 e23
    0b0111, 0b1011, 0b1101, 0b1110,          // e012 e013 e023 e123
    0b1111                                   // e0123
};

constexpr int popc4(int x) {
    int n = 0;
    while (x) { n += (x & 1); x >>= 1; }
    return n;
}

constexpr int idx_of(int mask) {
    for (int i = 0; i < 16; ++i)
        if (MASK[i] == mask) return i;
    return 0;
}

// #inversions when merging sorted generator lists of a then b == bubble-sort
// swap count in the reference.
constexpr int reorder_sign(int a, int b) {
    int n = 0;
    a >>= 1;
    while (a) { n += popc4(a & b); a >>= 1; }
    return (n & 1) ? -1 : 1;
}

struct Tab {
    int   kk[16][16];   // kk[j][i]: the unique k with blade_j*blade_k ~ blade_i
    float cf[16][16];   // sign(e_j*e_k) * rev(k), or 0 if e0^2 kills it
};

constexpr Tab make() {
    Tab t{};
    for (int j = 0; j < 16; ++j) {
        for (int i = 0; i < 16; ++i) {
            const int mj = MASK[j];
            const int mi = MASK[i];
            const int mk = mj ^ mi;           // group property: k is determined
            const int k  = idx_of(mk);
            t.kk[j][i] = k;
            float s = 0.0f;
            if (!(mj & mk & 1)) {             // shared e0 -> product vanishes
                const int g   = popc4(mk);
                const int rev = ((g * (g - 1) / 2) & 1) ? -1 : 1;
                s = (float)(reorder_sign(mj, mk) * rev);
            }
            t.cf[j][i] = s;
        }
    }
    return t;
}

} // namespace pga

__constant__ pga::Tab c_tab = pga::make();

// ---------------------------------------------------------------------------
// Kernel 1: build B[j, m*16 + i] = gp[i,j,k]*rev[k]*T[m,k]   (16 x 16N, 2 MB)
// One thread per (j, m); writes one contiguous 64B row (wave writes 2 KB).
// ---------------------------------------------------------------------------
__global__ void build_B(const float* __restrict__ T, float* __restrict__ B, int N) {
    const int tid   = blockIdx.x * blockDim.x + threadIdx.x;
    const int total = 16 * N;
    if (tid >= total) return;
    const int j = tid / N;
    const int m = tid - j * N;
    const size_t LDB = (size_t)N * 16;

    float w[16];
#pragma unroll
    for (int i = 0; i < 16; ++i) {
        const int   k = c_tab.kk[j][i];
        const float c = c_tab.cf[j][i];
        w[i] = c * T[(size_t)m * 16 + k];   // scattered 4B loads, L1-resident row
    }

    float4* dst = (float4*)(B + (size_t)j * LDB + (size_t)m * 16);
    dst[0] = float4{w[0],  w[1],  w[2],  w[3]};
    dst[1] = float4{w[4],  w[5],  w[6],  w[7]};
    dst[2] = float4{w[8],  w[9],  w[10], w[11]};
    dst[3] = float4{w[12], w[13], w[14], w[15]};
}

// ---------------------------------------------------------------------------
// Kernel 2: out(2048 x 16N) = T(2048 x 16) @ B(16 x 16N) via WMMA f32 16x16x4.
// One 16x16 C tile per wave iteration; 8 waves/block share the A n-tile.
// Specialized on compile-time CCOLS (32768 for the N=2048 problem) so every
// row stride folds into the 24-bit ioffset: all loads/stores become
// saddr + one shared 32-bit voffset + immediate — zero per-tile address VALU.
// CCOLS==0 instantiates a dynamic-width fallback.
// ---------------------------------------------------------------------------
typedef __attribute__((ext_vector_type(2))) float v2f;
typedef __attribute__((ext_vector_type(8))) float v8f;

#define WAVES_PER_BLOCK 8
#define TILES_PER_WAVE  4

template <unsigned CCOLS>
__global__ void __launch_bounds__(32 * WAVES_PER_BLOCK)
gemm_wmma(const float* __restrict__ T, const float* __restrict__ B,
          float* __restrict__ out, int N) {
    const unsigned cols = CCOLS ? CCOLS : (unsigned)N * 16u;   // matrix width
    const int lane = threadIdx.x & 31;
    // Pin wave id into an SGPR: tile indices become scalar, so the tile-bound
    // guard is a scalar branch and EXEC stays all-ones around WMMA.
    const int wave = __builtin_amdgcn_readfirstlane(threadIdx.x >> 5);
    const int half = lane >> 4;        // lane half selects K+0 / K+2 per A VGPR pair
    const int r    = lane & 15;        // row (A) or column (B/C) within tile
    const int n0   = blockIdx.y * 16;  // n-tile base row

    // ---- A tile (16 rows x K=16), ISA 32-bit A 16x4 layout per chunk:
    //      VGPR 2c: K = 4c + 2*half ; VGPR 2c+1: K = 4c + 2*half + 1
    const float*   abase = T + (size_t)n0 * 16;              // uniform
    const unsigned aoff  = (unsigned)(r * 16 + 2 * half);    // per-lane
    v2f a[4];
#pragma unroll
    for (int c = 0; c < 4; ++c)
        a[c] = *(const v2f*)(abase + aoff + 4u * c);

    // Per-lane offsets reused for every tile (bases advance by scalar 16/tile).
    const unsigned boff = (unsigned)(2 * half) * cols + (unsigned)r;       // B
    const unsigned ooff = (unsigned)(8 * half) * cols + (unsigned)r;       // C

    const int ct0 = (blockIdx.x * WAVES_PER_BLOCK + wave) * TILES_PER_WAVE;

    for (int t = 0; t < TILES_PER_WAVE; ++t) {         // scalar control flow
        const int ct = ct0 + t;
        const unsigned col0 = (unsigned)ct * 16u;
        if (col0 >= cols) break;                       // scalar guard

        const float* bbase = B + col0;                              // uniform
        float*       obase = out + (size_t)n0 * cols + col0;        // uniform

        // ---- B tile (K=16 x 16 cols): VGPR 2c row K=4c+2h, VGPR 2c+1 row +1.
        // Row strides (4c)*cols*4B (<2MB) are compile-time -> fold into ioffset.
        v2f b[4];
#pragma unroll
        for (int c = 0; c < 4; ++c) {
            b[c].x = bbase[boff + (unsigned)(4 * c) * cols];
            b[c].y = bbase[boff + (unsigned)(4 * c + 1) * cols];
        }

        v8f acc = {};
#pragma unroll
        for (int c = 0; c < 4; ++c) {
            // v_wmma_f32_16x16x4_f32 : D = A(16x4) * B(4x16) + C
            acc = __builtin_amdgcn_wmma_f32_16x16x4_f32(
                false, a[c], false, b[c], (short)0, acc, false, false);
        }

        // ---- C layout: VGPR vr holds row (vr + 8*half), column r. Each
        // 16-lane half writes one contiguous 64B line -> coalesced. NT hint:
        // the 268MB stream must not evict the 2MB B matrix from L2.
        // Row stride vr*cols*4B (<1MB) is compile-time -> folds into ioffset.
#pragma unroll
        for (int vr = 0; vr < 8; ++vr) {
            __builtin_nontemporal_store(
                acc[vr], obase + ooff + (unsigned)vr * cols);
        }
    }
}

extern "C" void kernel_launch(void* const* d_in, const int* in_sizes, int n_in,
                              void* d_out, int out_size, void* d_ws, size_t ws_size,
                              hipStream_t stream) {
    const float* T = (const float*)d_in[0];
    float* out     = (float*)d_out;
    float* B       = (float*)d_ws;              // needs 16 * 16N * 4 B (2 MB @ N=2048)
    const int N    = in_sizes[0] / 16;          // number of multivectors (2048)

    // Stage 1: build the 16 x 16N B matrix.
    {
        const int total   = 16 * N;
        const int threads = 256;
        build_B<<<(total + threads - 1) / threads, threads, 0, stream>>>(T, B, N);
    }

    // Stage 2: WMMA GEMM, out = T @ B.
    {
        const int CT = N;                        // column tiles (16 cols each)
        const int NT = N / 16;                   // row tiles
        const int tiles_per_block = WAVES_PER_BLOCK * TILES_PER_WAVE;
        dim3 grid((CT + tiles_per_block - 1) / tiles_per_block, NT);
        if (N == 2048) {
            gemm_wmma<2048u * 16u>
                <<<grid, 32 * WAVES_PER_BLOCK, 0, stream>>>(T, B, out, N);
        } else {
            gemm_wmma<0u>
                <<<grid, 32 * WAVES_PER_BLOCK, 0, stream>>>(T, B, out, N);
        }
    }
}